// Mamba_4569845202965
// MI455X (gfx1250) — compile-verified
//
#include <hip/hip_runtime.h>

// ---------------- problem constants ----------------
#define L_SEQ   4096
#define DMODEL  1024
#define DINNER  2048
#define E2      4096      // 2*DINNER
#define DTRANK  64
#define NSTATE  8
#define KPROJ   80        // DTRANK + 2*NSTATE
#define CHUNK   128
#define NCH     32        // L_SEQ / CHUNK

typedef __attribute__((ext_vector_type(2))) float v2f;
typedef __attribute__((ext_vector_type(8))) float v8f;
typedef __attribute__((ext_vector_type(4))) int   v4i;

__device__ __forceinline__ float siluf(float x)     { return x / (1.f + __expf(-x)); }
__device__ __forceinline__ float softplusf(float x) { return (x > 20.f) ? x : __logf(1.f + __expf(x)); }

// ---------------- CDNA5 async global->LDS path (hedged) ----------------
#if defined(__has_builtin)
#if __has_builtin(__builtin_amdgcn_global_load_async_to_lds_b128)
#define HAVE_ASYNC_LDS 1
#endif
#endif
#ifndef HAVE_ASYNC_LDS
#define HAVE_ASYNC_LDS 0
#endif

#if HAVE_ASYNC_LDS
// signature (from clang diagnostic): (v4i addrspace(1)*, v4i addrspace(3)*, imm offset, imm cpol)
#define ASYNC_CP16(gp, lp)                                                    \
    __builtin_amdgcn_global_load_async_to_lds_b128(                           \
        (__attribute__((address_space(1))) v4i*)(gp),                         \
        (__attribute__((address_space(3))) v4i*)(lp), 0, 0)
#if __has_builtin(__builtin_amdgcn_s_wait_asynccnt)
#define WAIT_ASYNC() __builtin_amdgcn_s_wait_asynccnt(0)
#else
#define WAIT_ASYNC() asm volatile("s_wait_asynccnt 0x0" ::: "memory")
#endif
#endif

// ========================================================================
// Generic WMMA fp32 GEMM:  C[M x N] = A[M x K] * B[N x K]^T   (both K-major)
// Block tile 128x64, BK=16, 8 waves (4x2), each wave 2x2 x (16x16) subtiles.
// Interior tiles: double-buffered GLOBAL_LOAD_ASYNC_TO_LDS_B128 (ASYNCcnt)
// overlapping the fill of tile k+1 with V_WMMA_F32_16X16X4_F32 on tile k.
// ========================================================================
#define BM 128
#define BN 64
#define BK 16
#define LDSK 20   // padded row stride (floats): 16B-aligned, conflict-free

__global__ __launch_bounds__(256)
void wmma_gemm_nt(const float* __restrict__ Ag, const float* __restrict__ Bg,
                  float* __restrict__ Cg,
                  int M, int N, int K, int lda, int ldb, int ldc,
                  long sA, long sB, long sC)
{
    __shared__ float As[2][BM * LDSK];
    __shared__ float Bs[2][BN * LDSK];

    const int zb = blockIdx.z;
    const float* A = Ag + (long)zb * sA;
    const float* B = Bg + (long)zb * sB;
    float*       C = Cg + (long)zb * sC;

    const int tid  = threadIdx.x;
    const int lane = tid & 31;
    const int wave = tid >> 5;
    const int wm   = wave & 3;        // 4 waves along M
    const int wn   = wave >> 2;       // 2 waves along N
    const int half = lane >> 4;       // K-pair selector (ISA A/B layout)
    const int lr   = lane & 15;       // row (A) / col (B/C) within 16
    const int m0   = blockIdx.y * BM;
    const int n0   = blockIdx.x * BN;

    v8f acc[2][2] = {};               // 4 x (16x16 f32) accumulators

    const int ldr  = tid >> 2;        // 0..63  (tile row for staging)
    const int ldq  = (tid & 3) * 4;   // 0,4,8,12 (float4 column)

    // ---- WMMA over one staged K-tile ----
    auto compute_tile = [&](const float* as, const float* bs) {
        #pragma unroll
        for (int kk = 0; kk < BK; kk += 4) {
            const int ks = kk + half * 2;   // lanes 0-15: K,K+1 ; lanes 16-31: K+2,K+3
            v2f a0 = *(const v2f*)(&as[(wm * 32 +      lr) * LDSK + ks]);
            v2f a1 = *(const v2f*)(&as[(wm * 32 + 16 + lr) * LDSK + ks]);
            v2f b0 = *(const v2f*)(&bs[(wn * 32 +      lr) * LDSK + ks]);
            v2f b1 = *(const v2f*)(&bs[(wn * 32 + 16 + lr) * LDSK + ks]);
            acc[0][0] = __builtin_amdgcn_wmma_f32_16x16x4_f32(false, a0, false, b0, (short)0, acc[0][0], false, false);
            acc[0][1] = __builtin_amdgcn_wmma_f32_16x16x4_f32(false, a0, false, b1, (short)0, acc[0][1], false, false);
            acc[1][0] = __builtin_amdgcn_wmma_f32_16x16x4_f32(false, a1, false, b0, (short)0, acc[1][0], false, false);
            acc[1][1] = __builtin_amdgcn_wmma_f32_16x16x4_f32(false, a1, false, b1, (short)0, acc[1][1], false, false);
        }
    };

    // ---- synchronous (guarded, zero-filled) staging for edge tiles ----
    auto stage_sync = [&](int k0, int buf) {
        #pragma unroll
        for (int rr = 0; rr < BM; rr += 64) {
            int gm = m0 + ldr + rr;
            float4 v = make_float4(0.f, 0.f, 0.f, 0.f);
            if (gm < M) v = *(const float4*)(A + (long)gm * lda + (k0 + ldq));
            *(float4*)(&As[buf][(ldr + rr) * LDSK + ldq]) = v;
        }
        int gn = n0 + ldr;
        float4 v = make_float4(0.f, 0.f, 0.f, 0.f);
        if (gn < N) v = *(const float4*)(B + (long)gn * ldb + (k0 + ldq));
        *(float4*)(&Bs[buf][ldr * LDSK + ldq]) = v;
    };

    const int nk = K / BK;            // all launches use K % BK == 0
    const bool interior = (m0 + BM <= M) && (n0 + BN <= N);

#if HAVE_ASYNC_LDS
    if (interior) {
        // ---- async staging: one b128 per lane per row-group, no VGPR bounce ----
        auto stage_async = [&](int k0, int buf) {
            #pragma unroll
            for (int rr = 0; rr < BM; rr += 64) {
                ASYNC_CP16(A + (long)(m0 + ldr + rr) * lda + (k0 + ldq),
                           &As[buf][(ldr + rr) * LDSK + ldq]);
            }
            ASYNC_CP16(B + (long)(n0 + ldr) * ldb + (k0 + ldq),
                       &Bs[buf][ldr * LDSK + ldq]);
        };

        stage_async(0, 0);
        WAIT_ASYNC();
        __syncthreads();
        for (int ki = 0; ki < nk; ++ki) {
            if (ki + 1 < nk) stage_async((ki + 1) * BK, (ki + 1) & 1);
            if (ki + 2 < nk) {  // L2 prefetch two tiles ahead (global_prefetch_b8)
                __builtin_prefetch(A + (long)(m0 + ldr) * lda + ((ki + 2) * BK + ldq), 0, 1);
                __builtin_prefetch(B + (long)(n0 + ldr) * ldb + ((ki + 2) * BK + ldq), 0, 1);
            }
            compute_tile(As[ki & 1], Bs[ki & 1]);
            WAIT_ASYNC();       // next tile's async copies have landed in LDS
            __syncthreads();    // all waves' copies visible / reads of cur done
        }
    } else
#endif
    {
        for (int ki = 0; ki < nk; ++ki) {
            stage_sync(ki * BK, 0);
            if (ki + 1 < nk) {
                __builtin_prefetch(A + (long)(m0 + ldr) * lda + ((ki + 1) * BK + ldq), 0, 1);
                __builtin_prefetch(B + (long)(n0 + ldr) * ldb + ((ki + 1) * BK + ldq), 0, 1);
            }
            __syncthreads();
            compute_tile(As[0], Bs[0]);
            __syncthreads();
        }
    }

    // ---- store: C VGPR r <-> row r (lanes 0-15) / r+8 (lanes 16-31) ----
    #pragma unroll
    for (int i = 0; i < 2; ++i) {
        #pragma unroll
        for (int j = 0; j < 2; ++j) {
            int mt = m0 + wm * 32 + i * 16 + half * 8;
            int nt = n0 + wn * 32 + j * 16 + lr;
            if (nt < N) {
                #pragma unroll
                for (int r = 0; r < 8; ++r) {
                    int m = mt + r;
                    if (m < M) C[(long)m * ldc + nt] = acc[i][j][r];
                }
            }
        }
    }
}

// ========================================================================
// Depthwise conv(4)+bias+SiLU (or plain SiLU) fused with (b,e,l)->(b,l,d)
// transpose via 32x32 LDS tile.
// ========================================================================
__global__ __launch_bounds__(256)
void conv_transpose_silu(const float* __restrict__ xz,
                         const float* __restrict__ conv_w,
                         const float* __restrict__ conv_b,
                         float* __restrict__ out_bld,
                         int eoff, int doConv)
{
    __shared__ float tile[32][33];
    const int b  = blockIdx.z;
    const int d0 = blockIdx.y * 32;
    const int l0 = blockIdx.x * 32;
    const int tl = threadIdx.x & 31;
    const int td = threadIdx.x >> 5;   // 8 rows per pass

    const float* base = xz + ((long)b * E2 + eoff) * (long)L_SEQ;

    #pragma unroll
    for (int p = 0; p < 32; p += 8) {
        const int d = d0 + td + p;
        const int l = l0 + tl;
        const float* row = base + (long)d * L_SEQ;
        float v;
        if (doConv) {
            float acc = conv_b[d];
            #pragma unroll
            for (int j = 0; j < 4; ++j) {
                int ll = l - 3 + j;
                float xv = (ll >= 0) ? row[ll] : 0.f;
                acc += xv * conv_w[d * 4 + j];
            }
            v = acc;
        } else {
            v = row[l];
        }
        tile[td + p][tl] = siluf(v);
    }
    __syncthreads();
    #pragma unroll
    for (int p = 0; p < 32; p += 8) {
        const int l = l0 + td + p;
        const int d = d0 + tl;
        out_bld[((long)b * L_SEQ + l) * DINNER + d] = tile[tl][td + p];
    }
}

// ========================================================================
// A[d][n] = -exp(A_log[d][n]) + scale * (U @ V)[d][n] / sqrt(8)
// ========================================================================
__global__ void build_A(const float* __restrict__ A_log, const float* __restrict__ U,
                        const float* __restrict__ V, const float* __restrict__ scale,
                        float* __restrict__ Amat)
{
    int t = blockIdx.x * 256 + threadIdx.x;
    if (t >= DINNER * NSTATE) return;
    int d = t >> 3, n = t & 7;
    float acc = 0.f;
    #pragma unroll
    for (int r = 0; r < 8; ++r) acc += U[d * 8 + r] * V[r * 8 + n];
    Amat[t] = -__expf(A_log[t]) + scale[0] * acc * 0.35355339059327373f; // 1/sqrt(8)
}

// ========================================================================
// Chunked parallel selective scan (linear recurrence):
//  pass A: per-chunk (prod dA, q) with h0=0, fully parallel over chunks
//  pass B: tiny sequential scan over chunk boundaries
//  pass C: replay chunks from correct h0, emit gated y in (b,l,d)
// ========================================================================
__global__ __launch_bounds__(256)
void scan_passA(const float* __restrict__ dtraw, const float* __restrict__ u_bld,
                const float* __restrict__ xdbl, const float* __restrict__ Amat,
                const float* __restrict__ dtb,
                float* __restrict__ Pout, float* __restrict__ qout)
{
    const int b  = blockIdx.z;
    const int ch = blockIdx.y;
    const int d  = blockIdx.x * 256 + threadIdx.x;
    __shared__ float sBC[CHUNK * 16];
    const long lbase = (long)b * L_SEQ + (long)ch * CHUNK;
    for (int i = threadIdx.x; i < CHUNK * 16; i += 256) {
        int s = i >> 4, c = i & 15;
        sBC[i] = xdbl[(lbase + s) * KPROJ + DTRANK + c];
    }
    __syncthreads();

    float Arow[8];
    #pragma unroll
    for (int n = 0; n < 8; ++n) Arow[n] = Amat[d * 8 + n];
    const float bias = dtb[d];

    float h[8] = {0.f,0.f,0.f,0.f,0.f,0.f,0.f,0.f};
    float P[8] = {1.f,1.f,1.f,1.f,1.f,1.f,1.f,1.f};

    for (int s = 0; s < CHUNK; ++s) {
        const long off = (lbase + s) * DINNER + d;
        const float delta = softplusf(dtraw[off] + bias);
        const float du    = delta * u_bld[off];
        const float* Bv   = &sBC[s * 16];
        #pragma unroll
        for (int n = 0; n < 8; ++n) {
            float dA = __expf(delta * Arow[n]);
            h[n] = h[n] * dA + du * Bv[n];
            P[n] *= dA;
        }
    }
    const long o = (((long)b * DINNER + d) * NCH + ch) * 8;
    #pragma unroll
    for (int n = 0; n < 8; ++n) { Pout[o + n] = P[n]; qout[o + n] = h[n]; }
}

__global__ void scan_passB(const float* __restrict__ P, const float* __restrict__ q,
                           float* __restrict__ hstart)
{
    const long t = blockIdx.x * 256 + threadIdx.x;   // b*DINNER + d, 4096 total
    float h[8] = {0.f,0.f,0.f,0.f,0.f,0.f,0.f,0.f};
    for (int c = 0; c < NCH; ++c) {
        const long o = (t * NCH + c) * 8;
        #pragma unroll
        for (int n = 0; n < 8; ++n) hstart[o + n] = h[n];
        #pragma unroll
        for (int n = 0; n < 8; ++n) h[n] = h[n] * P[o + n] + q[o + n];
    }
}

__global__ __launch_bounds__(256)
void scan_passC(const float* __restrict__ dtraw, const float* __restrict__ u_bld,
                const float* __restrict__ xdbl, const float* __restrict__ Amat,
                const float* __restrict__ dtb, const float* __restrict__ Dvec,
                const float* __restrict__ zsilu, const float* __restrict__ hstart,
                float* __restrict__ yg)
{
    const int b  = blockIdx.z;
    const int ch = blockIdx.y;
    const int d  = blockIdx.x * 256 + threadIdx.x;
    __shared__ float sBC[CHUNK * 16];
    const long lbase = (long)b * L_SEQ + (long)ch * CHUNK;
    for (int i = threadIdx.x; i < CHUNK * 16; i += 256) {
        int s = i >> 4, c = i & 15;
        sBC[i] = xdbl[(lbase + s) * KPROJ + DTRANK + c];
    }
    __syncthreads();

    float Arow[8];
    #pragma unroll
    for (int n = 0; n < 8; ++n) Arow[n] = Amat[d * 8 + n];
    const float bias = dtb[d];
    const float Dd   = Dvec[d];

    float h[8];
    const long o0 = (((long)b * DINNER + d) * NCH + ch) * 8;
    #pragma unroll
    for (int n = 0; n < 8; ++n) h[n] = hstart[o0 + n];

    for (int s = 0; s < CHUNK; ++s) {
        const long off = (lbase + s) * DINNER + d;
        const float u     = u_bld[off];
        const float delta = softplusf(dtraw[off] + bias);
        const float du    = delta * u;
        const float* Bv   = &sBC[s * 16];
        const float* Cv   = Bv + 8;
        float y = 0.f;
        #pragma unroll
        for (int n = 0; n < 8; ++n) {
            float dA = __expf(delta * Arow[n]);
            h[n] = h[n] * dA + du * Bv[n];
            y += h[n] * Cv[n];
        }
        y += u * Dd;
        yg[off] = y * zsilu[off];
    }
}

// ========================================================================
extern "C" void kernel_launch(void* const* d_in, const int* in_sizes, int n_in,
                              void* d_out, int out_size, void* d_ws, size_t ws_size,
                              hipStream_t stream)
{
    const float* hidden     = (const float*)d_in[0];
    const float* in_proj_w  = (const float*)d_in[1];
    const float* conv_w     = (const float*)d_in[2];
    const float* conv_b     = (const float*)d_in[3];
    const float* x_proj_w   = (const float*)d_in[4];
    const float* dt_proj_w  = (const float*)d_in[5];
    const float* dt_proj_b  = (const float*)d_in[6];
    const float* A_log      = (const float*)d_in[7];
    const float* Dv         = (const float*)d_in[8];
    const float* out_proj_w = (const float*)d_in[9];
    const float* U          = (const float*)d_in[10];
    const float* V          = (const float*)d_in[11];
    const float* scale      = (const float*)d_in[12];
    float* out = (float*)d_out;

    char* ws = (char*)d_ws;
    size_t off = 0;
    auto alloc = [&](size_t bytes) -> float* {
        float* p = (float*)(ws + off);
        off += (bytes + 255) & ~(size_t)255;
        return p;
    };
    float* xz     = alloc((size_t)2 * E2 * L_SEQ * 4);        // (b,e,l)
    float* x_bld  = alloc((size_t)2 * L_SEQ * DINNER * 4);    // conv+silu, (b,l,d)
    float* zsilu  = alloc((size_t)2 * L_SEQ * DINNER * 4);    // silu(z), (b,l,d)
    float* xdbl   = alloc((size_t)2 * L_SEQ * KPROJ * 4);     // (b,l,80)
    float* dtraw  = alloc((size_t)2 * L_SEQ * DINNER * 4);    // pre-softplus dt
    float* Amat   = alloc((size_t)DINNER * NSTATE * 4);
    float* Pbuf   = alloc((size_t)2 * DINNER * NCH * 8 * 4);
    float* qbuf   = alloc((size_t)2 * DINNER * NCH * 8 * 4);
    float* hstart = alloc((size_t)2 * DINNER * NCH * 8 * 4);
    float* yg     = alloc((size_t)2 * L_SEQ * DINNER * 4);    // gated y, (b,l,d)

    dim3 blk(256);

    // 1) in_proj: xz[b][e][l] = W_in[e][:] . hidden[b][l][:]
    wmma_gemm_nt<<<dim3(L_SEQ / BN, E2 / BM, 2), blk, 0, stream>>>(
        in_proj_w, hidden, xz, E2, L_SEQ, DMODEL,
        DMODEL, DMODEL, L_SEQ,
        0L, (long)L_SEQ * DMODEL, (long)E2 * L_SEQ);

    // 2) conv+SiLU+transpose (x half), SiLU+transpose (z half)
    conv_transpose_silu<<<dim3(L_SEQ / 32, DINNER / 32, 2), blk, 0, stream>>>(
        xz, conv_w, conv_b, x_bld, 0, 1);
    conv_transpose_silu<<<dim3(L_SEQ / 32, DINNER / 32, 2), blk, 0, stream>>>(
        xz, conv_w, conv_b, zsilu, DINNER, 0);

    // 3) A matrix (tiny)
    build_A<<<dim3((DINNER * NSTATE + 255) / 256), blk, 0, stream>>>(A_log, U, V, scale, Amat);

    // 4) x_proj: xdbl[b][l][k] = x_bld[b][l][:] . x_proj_w[k][:]
    wmma_gemm_nt<<<dim3((KPROJ + BN - 1) / BN, L_SEQ / BM, 2), blk, 0, stream>>>(
        x_bld, x_proj_w, xdbl, L_SEQ, KPROJ, DINNER,
        DINNER, DINNER, KPROJ,
        (long)L_SEQ * DINNER, 0L, (long)L_SEQ * KPROJ);

    // 5) dt: dtraw[b][l][d] = xdbl[b][l][0:64] . dt_proj_w[d][:]
    wmma_gemm_nt<<<dim3(DINNER / BN, L_SEQ / BM, 2), blk, 0, stream>>>(
        xdbl, dt_proj_w, dtraw, L_SEQ, DINNER, DTRANK,
        KPROJ, DTRANK, DINNER,
        (long)L_SEQ * KPROJ, 0L, (long)L_SEQ * DINNER);

    // 6) chunked parallel scan + gating
    scan_passA<<<dim3(DINNER / 256, NCH, 2), blk, 0, stream>>>(
        dtraw, x_bld, xdbl, Amat, dt_proj_b, Pbuf, qbuf);
    scan_passB<<<dim3((2 * DINNER) / 256), blk, 0, stream>>>(Pbuf, qbuf, hstart);
    scan_passC<<<dim3(DINNER / 256, NCH, 2), blk, 0, stream>>>(
        dtraw, x_bld, xdbl, Amat, dt_proj_b, Dv, zsilu, hstart, yg);

    // 7) out_proj: out[b][l][m] = yg[b][l][:] . out_proj_w[m][:]
    wmma_gemm_nt<<<dim3(DMODEL / BN, L_SEQ / BM, 2), blk, 0, stream>>>(
        yg, out_proj_w, out, L_SEQ, DMODEL, DINNER,
        DINNER, DINNER, DMODEL,
        (long)L_SEQ * DINNER, 0L, (long)L_SEQ * DMODEL);
}